// GCN_64484638982596
// MI455X (gfx1250) — compile-verified
//
#include <hip/hip_runtime.h>
#include <math.h>

// GCN forward for MI455X (gfx1250, wave32).
// fp32 throughout; matmuls via V_WMMA_F32_16X16X4_F32 (exact fp32 semantics —
// workload is bandwidth-bound, ~100 MB HBM + L2-resident scatter, so the fp32
// WMMA costs nothing vs bf16 and matches the reference numerically).
//
// d_ws layout (floats): dinv[N] | h1[N*48] | agg1[N*48] | h2[N*16]  (~45.3 MB)

#define IN_F  128
#define HID_F 48
#define OUT_F 16

typedef float v2f __attribute__((ext_vector_type(2)));
typedef float v8f __attribute__((ext_vector_type(8)));

// ---------------------------------------------------------------- degree / norm
__global__ void k_init_deg(float* __restrict__ deg, int N) {
    int i = blockIdx.x * blockDim.x + threadIdx.x;
    if (i < N) deg[i] = 1.0f;              // self-loop contributes 1
}

__global__ void k_count_deg(const int* __restrict__ dst, float* __restrict__ deg, int E) {
    int e = blockIdx.x * blockDim.x + threadIdx.x;
    if (e < E) atomicAdd(&deg[dst[e]], 1.0f);
}

__global__ void k_rsqrt(float* __restrict__ deg, int N) {
    int i = blockIdx.x * blockDim.x + threadIdx.x;
    if (i < N) deg[i] = rsqrtf(deg[i]);    // deg >= 1 always (self-loops)
}

// ---------------------------------------------------------------- GEMM1: h1 = X @ W1
// block = 128 threads (4 wave32); block computes a 64-row x 48-col tile.
// fp32 WMMA fragment layouts (CDNA5 ISA 7.12.2):
//   A 16x4 : lane l -> row M=l%16 ; VGPR0/1 hold K = (l/16)*2 + {0,1}
//   B 4x16 : lane l -> col N=l%16 ; VGPR0/1 hold K = (l/16)*2 + {0,1}
//   C 16x16: element r of v8f -> row M = r + (l/16)*8, col N = l%16
// W is staged TRANSPOSED in LDS so both A and B fragments are single ds_load_b64.
__global__ __launch_bounds__(128) void k_gemm1(const float* __restrict__ X,
                                               const float* __restrict__ W1,
                                               float* __restrict__ H, int N) {
    __shared__ float sX [64 * IN_F];     // 32 KB, row-major
    __shared__ float sWt[HID_F * IN_F];  // 24 KB, TRANSPOSED: sWt[c*128 + k]
    const int tid  = threadIdx.x;
    const int wave = tid >> 5;
    const int lane = tid & 31;
    const int rowBase = blockIdx.x * 64;

    // stage W1 transposed (coalesced float4 reads, scattered LDS writes; one-time)
    for (int i = tid; i < IN_F * HID_F / 4; i += 128) {
        int k = i / 12, c4 = (i % 12) * 4;
        float4 w = *(const float4*)&W1[k * HID_F + c4];
        sWt[(c4 + 0) * IN_F + k] = w.x;
        sWt[(c4 + 1) * IN_F + k] = w.y;
        sWt[(c4 + 2) * IN_F + k] = w.z;
        sWt[(c4 + 3) * IN_F + k] = w.w;
    }
    // stage X tile 64x128 (coalesced float4, zero-fill OOB rows)
    for (int i = tid; i < 64 * IN_F / 4; i += 128) {
        int r = i >> 5, c4 = i & 31;
        int gr = rowBase + r;
        float4 v = {0.f, 0.f, 0.f, 0.f};
        if (gr < N) v = *(const float4*)&X[(size_t)gr * IN_F + c4 * 4];
        *(float4*)&sX[r * IN_F + c4 * 4] = v;
    }
    __syncthreads();

    const int mrow  = lane & 15;
    const int khalf = (lane >> 4) * 2;
    const float* aRow = &sX[(wave * 16 + mrow) * IN_F + khalf];

    v8f acc[3] = {};
    for (int k = 0; k < IN_F; k += 4) {
        const v2f a = *(const v2f*)&aRow[k];
#pragma unroll
        for (int nt = 0; nt < 3; ++nt) {
            const int col = nt * 16 + mrow;
            const v2f b = *(const v2f*)&sWt[col * IN_F + k + khalf];
            acc[nt] = __builtin_amdgcn_wmma_f32_16x16x4_f32(
                false, a, false, b, (short)0, acc[nt], false, false);
        }
    }

    const int rowOff = rowBase + wave * 16 + (lane >> 4) * 8;
    if (rowBase + 64 <= N) {               // fast path: block-uniform, no lane guards
#pragma unroll
        for (int nt = 0; nt < 3; ++nt)
#pragma unroll
            for (int r = 0; r < 8; ++r)
                H[(size_t)(rowOff + r) * HID_F + nt * 16 + mrow] = acc[nt][r];
    } else {
#pragma unroll
        for (int nt = 0; nt < 3; ++nt)
#pragma unroll
            for (int r = 0; r < 8; ++r)
                if (rowOff + r < N)
                    H[(size_t)(rowOff + r) * HID_F + nt * 16 + mrow] = acc[nt][r];
    }
}

// ---------------------------------------------------------------- GEMM2: h2 = relu(agg1) @ W2
__global__ __launch_bounds__(128) void k_gemm2(const float* __restrict__ A1,
                                               const float* __restrict__ W2,
                                               float* __restrict__ H2, int N) {
    __shared__ float sA [64 * HID_F];     // 12 KB, row-major
    __shared__ float sWt[OUT_F * HID_F];  // 3 KB, TRANSPOSED: sWt[c*48 + k]
    const int tid  = threadIdx.x;
    const int wave = tid >> 5;
    const int lane = tid & 31;
    const int rowBase = blockIdx.x * 64;

    for (int i = tid; i < HID_F * OUT_F / 4; i += 128) {
        int k = i >> 2, c4 = (i & 3) * 4;
        float4 w = *(const float4*)&W2[k * OUT_F + c4];
        sWt[(c4 + 0) * HID_F + k] = w.x;
        sWt[(c4 + 1) * HID_F + k] = w.y;
        sWt[(c4 + 2) * HID_F + k] = w.z;
        sWt[(c4 + 3) * HID_F + k] = w.w;
    }
    for (int i = tid; i < 64 * HID_F / 4; i += 128) {
        int r = i / 12, c4 = i % 12;
        int gr = rowBase + r;
        float4 v = {0.f, 0.f, 0.f, 0.f};
        if (gr < N) {
            v = *(const float4*)&A1[(size_t)gr * HID_F + c4 * 4];
            v.x = fmaxf(v.x, 0.f); v.y = fmaxf(v.y, 0.f);   // fused ReLU
            v.z = fmaxf(v.z, 0.f); v.w = fmaxf(v.w, 0.f);
        }
        *(float4*)&sA[r * HID_F + c4 * 4] = v;
    }
    __syncthreads();

    const int mrow  = lane & 15;
    const int khalf = (lane >> 4) * 2;
    const float* aRow = &sA[(wave * 16 + mrow) * HID_F + khalf];
    const float* bCol = &sWt[mrow * HID_F + khalf];

    v8f acc = {};
#pragma unroll
    for (int k = 0; k < HID_F; k += 4) {
        const v2f a = *(const v2f*)&aRow[k];
        const v2f b = *(const v2f*)&bCol[k];
        acc = __builtin_amdgcn_wmma_f32_16x16x4_f32(
            false, a, false, b, (short)0, acc, false, false);
    }

    const int rowOff = rowBase + wave * 16 + (lane >> 4) * 8;
    if (rowBase + 64 <= N) {
#pragma unroll
        for (int r = 0; r < 8; ++r)
            H2[(size_t)(rowOff + r) * OUT_F + mrow] = acc[r];
    } else {
#pragma unroll
        for (int r = 0; r < 8; ++r)
            if (rowOff + r < N)
                H2[(size_t)(rowOff + r) * OUT_F + mrow] = acc[r];
    }
}

// ---------------------------------------------------------------- bias init + scatter
__global__ void k_init_bias(float* __restrict__ out, const float* __restrict__ b,
                            int total, int F) {
    int i = blockIdx.x * blockDim.x + threadIdx.x;
    if (i < total) out[i] = b[i % F];
}

// One thread per (edge | self-loop); unrolled loop over F/4 float4 chunks.
// Loads src/dst/dinv once per edge; h[src] gather is 192B (L1=48) contiguous
// per lane; scatter is GLOBAL_ATOMIC_ADD_F32 into L2-resident agg.
template <int F>
__global__ void k_aggregate(const float* __restrict__ h, const float* __restrict__ dinv,
                            const int* __restrict__ srcA, const int* __restrict__ dstA,
                            float* __restrict__ agg, int E, int N) {
    int e = blockIdx.x * blockDim.x + threadIdx.x;
    if (e >= E + N) return;
    int s, d;
    if (e < E) { s = srcA[e]; d = dstA[e]; }
    else       { s = d = e - E; }                 // self-loop
    const float nm = dinv[s] * dinv[d];
    const float4* __restrict__ hp = (const float4*)&h[(size_t)s * F];
    float* __restrict__ p = &agg[(size_t)d * F];
#pragma unroll
    for (int c = 0; c < F / 4; ++c) {
        const float4 v = hp[c];
        atomicAdd(p + 4 * c + 0, v.x * nm);
        atomicAdd(p + 4 * c + 1, v.y * nm);
        atomicAdd(p + 4 * c + 2, v.z * nm);
        atomicAdd(p + 4 * c + 3, v.w * nm);
    }
}

// ---------------------------------------------------------------- log_softmax (in place, 16 cols)
__global__ void k_logsoftmax(float* __restrict__ io, int N) {
    int i = blockIdx.x * blockDim.x + threadIdx.x;
    if (i >= N) return;
    float v[OUT_F];
    float mx = -INFINITY;
#pragma unroll
    for (int j = 0; j < OUT_F; ++j) { v[j] = io[(size_t)i * OUT_F + j]; mx = fmaxf(mx, v[j]); }
    float s = 0.f;
#pragma unroll
    for (int j = 0; j < OUT_F; ++j) s += expf(v[j] - mx);
    const float l = mx + logf(s);
#pragma unroll
    for (int j = 0; j < OUT_F; ++j) io[(size_t)i * OUT_F + j] = v[j] - l;
}

// ---------------------------------------------------------------- launcher
extern "C" void kernel_launch(void* const* d_in, const int* in_sizes, int n_in,
                              void* d_out, int out_size, void* d_ws, size_t ws_size,
                              hipStream_t stream) {
    const float* X  = (const float*)d_in[0];
    const int*   ei = (const int*)d_in[1];      // [2,E] flattened: src row then dst row
    const float* W1 = (const float*)d_in[2];
    const float* b1 = (const float*)d_in[3];
    const float* W2 = (const float*)d_in[4];
    const float* b2 = (const float*)d_in[5];

    const int N = in_sizes[0] / IN_F;
    const int E = in_sizes[1] / 2;
    const int* srcA = ei;
    const int* dstA = ei + E;

    float* ws   = (float*)d_ws;
    float* dinv = ws;                              // N
    float* h1   = dinv + N;                        // N*48
    float* agg1 = h1 + (size_t)N * HID_F;          // N*48
    float* h2   = agg1 + (size_t)N * HID_F;        // N*16
    float* outp = (float*)d_out;                   // N*16 (agg2 then log-probs)

    const int T = 256;
    // degrees -> dinv
    k_init_deg <<<(N + T - 1) / T, T, 0, stream>>>(dinv, N);
    k_count_deg<<<(E + T - 1) / T, T, 0, stream>>>(dstA, dinv, E);
    k_rsqrt    <<<(N + T - 1) / T, T, 0, stream>>>(dinv, N);

    // layer 1
    k_gemm1<<<(N + 63) / 64, 128, 0, stream>>>(X, W1, h1, N);
    k_init_bias<<<((size_t)N * HID_F + T - 1) / T, T, 0, stream>>>(agg1, b1, N * HID_F, HID_F);
    k_aggregate<HID_F><<<(E + N + T - 1) / T, T, 0, stream>>>(h1, dinv, srcA, dstA, agg1, E, N);

    // layer 2 (ReLU fused into GEMM2's A-tile load)
    k_gemm2<<<(N + 63) / 64, 128, 0, stream>>>(agg1, W2, h2, N);
    k_init_bias<<<((size_t)N * OUT_F + T - 1) / T, T, 0, stream>>>(outp, b2, N * OUT_F, OUT_F);
    k_aggregate<OUT_F><<<(E + N + T - 1) / T, T, 0, stream>>>(h2, dinv, srcA, dstA, outp, E, N);

    k_logsoftmax<<<(N + T - 1) / T, T, 0, stream>>>(outp, N);
}